// MultiHeadGATLayer_86251533238847
// MI455X (gfx1250) — compile-verified
//
#include <hip/hip_runtime.h>

typedef float v2f __attribute__((ext_vector_type(2)));
typedef float v8f __attribute__((ext_vector_type(8)));

#define IN_DIM    256
#define OUT_DIM   64
#define ZLD       128   // z row: [head0 0..63 | head1 0..63]
#define ROWS_BLK  32    // rows per block (two 16-row WMMA tiles sharing B)
#define LDS_PITCH 260   // 256 + 4 pad: 16B-aligned rows (1040B), conflict-free A reads

__device__ __forceinline__ unsigned enc_f32(float f) {
    unsigned u = __float_as_uint(f);
    return (u & 0x80000000u) ? ~u : (u | 0x80000000u);
}
__device__ __forceinline__ float dec_f32(unsigned u) {
    return (u & 0x80000000u) ? __uint_as_float(u & 0x7FFFFFFFu)
                             : __uint_as_float(~u);
}

// ---------------------------------------------------------------------------
// K0: zero out / init workspace accumulators
// ---------------------------------------------------------------------------
__global__ void gat_init(float* __restrict__ out, unsigned* __restrict__ m_enc,
                         float* __restrict__ denom, int n_out, int n_nd) {
    int i = blockIdx.x * 256 + threadIdx.x;
    if (i < n_out) out[i] = 0.0f;
    if (i < n_nd)  { m_enc[i] = 0u; denom[i] = 0.0f; }  // enc(x) > 0 for all finite x
}

// ---------------------------------------------------------------------------
// K1: z = h @ W  (both heads) via V_WMMA_F32_16X16X4_F32.
// Block = 256 thr (8 wave32). Block covers 32 rows; wave w covers combined
// column tile [16w, 16w+16). Each wave accumulates TWO 16x16 tiles (row tile 0
// and 1) that share every B fragment -> 2 WMMAs per W load pair.
// h tile (32x256 f32) staged via float4 into padded LDS.
// ---------------------------------------------------------------------------
__global__ void gat_gemm(const float* __restrict__ h, const float* __restrict__ W,
                         float* __restrict__ z, int N) {
    __shared__ float ldsA[ROWS_BLK * LDS_PITCH];
    const int tid = threadIdx.x;
    const int rb  = blockIdx.x * ROWS_BLK;
    const bool full = (rb + ROWS_BLK) <= N;

    // cooperative float4 stage of 32x256 h tile (64 float4 per row)
    if (full) {
        for (int idx = tid; idx < ROWS_BLK * (IN_DIM / 4); idx += 256) {
            int r = idx >> 6, c4 = idx & 63;
            float4 v = ((const float4*)(h + (long)(rb + r) * IN_DIM))[c4];
            *(float4*)&ldsA[r * LDS_PITCH + c4 * 4] = v;
        }
    } else {
        for (int idx = tid; idx < ROWS_BLK * (IN_DIM / 4); idx += 256) {
            int r = idx >> 6, c4 = idx & 63;
            int row = rb + r; if (row >= N) row = N - 1;
            float4 v = ((const float4*)(h + (long)row * IN_DIM))[c4];
            *(float4*)&ldsA[r * LDS_PITCH + c4 * 4] = v;
        }
    }
    __syncthreads();

    const int wave = tid >> 5, lane = tid & 31;
    const int g = lane >> 4, r16 = lane & 15;     // half-wave group / sub-lane
    const int c = wave * 16 + r16;                // combined output column
    const int head = c >> 6, j = c & 63;
    const float* Bp = W + head * (IN_DIM * OUT_DIM) + j;  // W[head][k][j]
    const float* A0 = &ldsA[r16 * LDS_PITCH];
    const float* A1 = &ldsA[(16 + r16) * LDS_PITCH];

    v8f acc0 = {}, acc1 = {};
    #pragma unroll 8
    for (int k0 = 0; k0 < IN_DIM; k0 += 4) {
        const int kk = k0 + 2 * g;  // lanes 0-15: K=k0,k0+1 ; lanes 16-31: K=k0+2,k0+3
        v2f b;
        b.x = Bp[kk * OUT_DIM];
        b.y = Bp[(kk + 1) * OUT_DIM];
        v2f a0, a1;
        a0.x = A0[kk]; a0.y = A0[kk + 1];
        a1.x = A1[kk]; a1.y = A1[kk + 1];
        acc0 = __builtin_amdgcn_wmma_f32_16x16x4_f32(
            false, a0, false, b, (short)0, acc0, false, false);
        acc1 = __builtin_amdgcn_wmma_f32_16x16x4_f32(
            false, a1, false, b, (short)0, acc1, false, false);
    }

    // D layout: lane gives N=r16; VGPR v gives M = v + 8*g
    if (full) {
        #pragma unroll
        for (int v = 0; v < 8; ++v) {
            z[(long)(rb + v + 8 * g) * ZLD + c]      = acc0[v];
            z[(long)(rb + 16 + v + 8 * g) * ZLD + c] = acc1[v];
        }
    } else {
        #pragma unroll
        for (int v = 0; v < 8; ++v) {
            int r0 = rb + v + 8 * g, r1 = r0 + 16;
            if (r0 < N) z[(long)r0 * ZLD + c] = acc0[v];
            if (r1 < N) z[(long)r1 * ZLD + c] = acc1[v];
        }
    }
}

// ---------------------------------------------------------------------------
// K2: per-node attention scalars s_src[n][h] = z_h(n) . a_src_h,
//                                 s_dst[n][h] = z_h(n) . a_dst_h
// One wave per node; shfl-xor tree reduction. a is (2,128): [src64|dst64]/head.
// ---------------------------------------------------------------------------
__global__ void gat_scores(const float* __restrict__ z, const float* __restrict__ a,
                           float* __restrict__ s_src, float* __restrict__ s_dst, int N) {
    const int lane = threadIdx.x & 31;
    const int n = blockIdx.x * 8 + (threadIdx.x >> 5);
    if (n >= N) return;
    const float* zr = z + (long)n * ZLD;
    float z0 = zr[lane], z1 = zr[32 + lane];        // head 0
    float z2 = zr[64 + lane], z3 = zr[96 + lane];   // head 1
    float ss0 = z0 * a[lane]       + z1 * a[32 + lane];
    float sd0 = z0 * a[64 + lane]  + z1 * a[96 + lane];
    float ss1 = z2 * a[128 + lane] + z3 * a[160 + lane];
    float sd1 = z2 * a[192 + lane] + z3 * a[224 + lane];
    #pragma unroll
    for (int off = 16; off > 0; off >>= 1) {
        ss0 += __shfl_xor(ss0, off);
        sd0 += __shfl_xor(sd0, off);
        ss1 += __shfl_xor(ss1, off);
        sd1 += __shfl_xor(sd1, off);
    }
    if (lane == 0) {
        s_src[n * 2 + 0] = ss0; s_dst[n * 2 + 0] = sd0;
        s_src[n * 2 + 1] = ss1; s_dst[n * 2 + 1] = sd1;
    }
}

// ---------------------------------------------------------------------------
// K3: e[k][h] = leaky_relu(s_src[src]+s_dst[dst]); segment max via u32 atomicMax
// ---------------------------------------------------------------------------
__global__ void gat_edge_max(const int* __restrict__ src, const int* __restrict__ dst,
                             const float* __restrict__ s_src, const float* __restrict__ s_dst,
                             float* __restrict__ e_buf, unsigned* __restrict__ m_enc, int E) {
    int k = blockIdx.x * 256 + threadIdx.x;
    if (k >= E) return;
    int s = src[k], d = dst[k];
    #pragma unroll
    for (int hh = 0; hh < 2; ++hh) {
        float e = s_src[s * 2 + hh] + s_dst[d * 2 + hh];
        e = (e > 0.0f) ? e : 0.01f * e;
        e_buf[(long)k * 2 + hh] = e;
        atomicMax(&m_enc[d * 2 + hh], enc_f32(e));
    }
}

// ---------------------------------------------------------------------------
// K4: ex = exp(e - m[dst]); denom[dst] += ex  (segment sum)
// ---------------------------------------------------------------------------
__global__ void gat_edge_exp(const int* __restrict__ dst, float* __restrict__ e_buf,
                             const unsigned* __restrict__ m_enc,
                             float* __restrict__ denom, int E) {
    int k = blockIdx.x * 256 + threadIdx.x;
    if (k >= E) return;
    int d = dst[k];
    #pragma unroll
    for (int hh = 0; hh < 2; ++hh) {
        float m  = dec_f32(m_enc[d * 2 + hh]);
        float ex = __expf(e_buf[(long)k * 2 + hh] - m);
        e_buf[(long)k * 2 + hh] = ex;
        atomicAdd(&denom[d * 2 + hh], ex);
    }
}

// ---------------------------------------------------------------------------
// K5: out[dst] += 0.5 * sum_h alpha_h * z_h[src]   (head mean fused, wave/edge)
// z gathers + output atomics are L2-resident (z=51MB, out=25.6MB, L2=192MB).
// ---------------------------------------------------------------------------
__global__ void gat_edge_agg(const int* __restrict__ src, const int* __restrict__ dst,
                             const float* __restrict__ e_buf, const float* __restrict__ denom,
                             const float* __restrict__ z, float* __restrict__ out, int E) {
    const int lane = threadIdx.x & 31;
    const int k = blockIdx.x * 8 + (threadIdx.x >> 5);
    if (k >= E) return;
    int s = src[k], d = dst[k];
    float den0 = denom[d * 2 + 0], den1 = denom[d * 2 + 1];
    float al0 = e_buf[(long)k * 2 + 0] / (den0 == 0.0f ? 1.0f : den0);
    float al1 = e_buf[(long)k * 2 + 1] / (den1 == 0.0f ? 1.0f : den1);
    const float* zr = z + (long)s * ZLD;
    float c0 = 0.5f * (al0 * zr[lane]      + al1 * zr[64 + lane]);  // col = lane
    float c1 = 0.5f * (al0 * zr[32 + lane] + al1 * zr[96 + lane]);  // col = lane+32
    atomicAdd(&out[(long)d * OUT_DIM + lane],      c0);
    atomicAdd(&out[(long)d * OUT_DIM + 32 + lane], c1);
}

// ---------------------------------------------------------------------------
extern "C" void kernel_launch(void* const* d_in, const int* in_sizes, int n_in,
                              void* d_out, int out_size, void* d_ws, size_t ws_size,
                              hipStream_t stream) {
    const float* h = (const float*)d_in[0];
    const float* W = (const float*)d_in[1];
    const float* a = (const float*)d_in[2];
    const int* src = (const int*)d_in[3];
    const int* dst = (const int*)d_in[4];
    float* out = (float*)d_out;

    const int N = in_sizes[0] / IN_DIM;   // 100000
    const int E = in_sizes[3];            // 1600000

    // workspace carve-up (256B aligned)
    char* ws = (char*)d_ws;
    size_t off = 0;
    auto take = [&](size_t bytes) { char* p = ws + off; off = (off + bytes + 255) & ~(size_t)255; return p; };
    float*    z      = (float*)   take((size_t)N * ZLD * sizeof(float));  // 51.2 MB
    float*    s_src  = (float*)   take((size_t)N * 2 * sizeof(float));
    float*    s_dst  = (float*)   take((size_t)N * 2 * sizeof(float));
    unsigned* m_enc  = (unsigned*)take((size_t)N * 2 * sizeof(unsigned));
    float*    denom  = (float*)   take((size_t)N * 2 * sizeof(float));
    float*    e_buf  = (float*)   take((size_t)E * 2 * sizeof(float));    // 12.8 MB

    const int n_out = N * OUT_DIM;
    const int n_nd  = N * 2;

    gat_init<<<(n_out + 255) / 256, 256, 0, stream>>>(out, m_enc, denom, n_out, n_nd);
    gat_gemm<<<(N + ROWS_BLK - 1) / ROWS_BLK, 256, 0, stream>>>(h, W, z, N);
    gat_scores<<<(N + 7) / 8, 256, 0, stream>>>(z, a, s_src, s_dst, N);
    gat_edge_max<<<(E + 255) / 256, 256, 0, stream>>>(src, dst, s_src, s_dst, e_buf, m_enc, E);
    gat_edge_exp<<<(E + 255) / 256, 256, 0, stream>>>(dst, e_buf, m_enc, denom, E);
    gat_edge_agg<<<(E + 7) / 8, 256, 0, stream>>>(src, dst, e_buf, denom, z, out, E);
}